// Unfolder_35897336660639
// MI455X (gfx1250) — compile-verified
//
#include <hip/hip_runtime.h>

// Unfold (im2col) for B=8, C=3, H=W=512, k=3, pad=1.
// out[b,c,3i+u,3j+r] = xpad[b,c, i+u, j+r]   (xpad = zero-padded 514x514)
//
// Pure data movement: ~226.5 MB streamed output + 25.2 MB input reads.
// Strategy: one block per padded input row s (0..513) per (b,c).
//  - Stage the 512 interior floats of the row into LDS via
//    GLOBAL_LOAD_ASYNC_TO_LDS_B64 (ASYNCcnt-tracked CDNA5 async path),
//    zero the 2 halo columns / boundary rows with DS stores.
//  - Emit up to 3 output rows (p = 3*(s-u)+u) with coalesced float4 stores.

#define BDIM   256
#define IN_W   512
#define PAD_W  514          // padded row length
#define OUT_W  1536         // 3 * 512
#define BC     24           // B * C

__global__ __launch_bounds__(BDIM) void unfold_rows_kernel(
    const float* __restrict__ x, float* __restrict__ out) {
  // row[t] == lbuf[t+1]; lbuf is 16B-aligned so row[1] (first interior
  // element) lands on an 8B-aligned LDS address for the b64 async copies.
  __shared__ __align__(16) float lbuf[PAD_W + 4];

  const int s   = blockIdx.x;   // padded row index 0..513
  const int bc  = blockIdx.y;   // 0..23
  const int tid = threadIdx.x;

  if (s >= 1 && s <= IN_W) {
    // Interior row: async-copy x[bc, s-1, 0..511] into row[1..512].
    const float* grow = x + ((size_t)bc * IN_W + (size_t)(s - 1)) * IN_W;
    const int t = tid * 2;  // 0..510, two floats per lane => 512 exactly
    unsigned lds_off = (unsigned)(unsigned long long)(const void*)&lbuf[2 + t];
    unsigned long long ga = (unsigned long long)(const void*)&grow[t];
    asm volatile("global_load_async_to_lds_b64 %0, %1, off"
                 :: "v"(lds_off), "v"(ga) : "memory");
    asm volatile("s_wait_asynccnt 0" ::: "memory");
    if (tid == 0) lbuf[1] = 0.0f;              // row[0]   (left pad col)
    if (tid == 1) lbuf[1 + (PAD_W - 1)] = 0.0f; // row[513] (right pad col)
  } else {
    // Boundary padded row: all zeros.
    for (int t = tid; t < PAD_W; t += BDIM) lbuf[1 + t] = 0.0f;
  }
  __syncthreads();

  const float* __restrict__ row = lbuf + 1;

  // Each padded input row s feeds output rows p = 3i+u with i+u == s.
  #pragma unroll
  for (int u = 0; u < 3; ++u) {
    const int i = s - u;
    if (i < 0 || i >= IN_W) continue;
    const int p = 3 * i + u;
    float4* __restrict__ orow =
        (float4*)(out + ((size_t)bc * OUT_W + (size_t)p) * OUT_W);
    // 1536 floats = 384 float4 per row; coalesced streaming stores.
    #pragma unroll
    for (int base = 0; base < OUT_W / 4; base += BDIM) {
      const int idx = base + tid;
      if (idx < OUT_W / 4) {
        const int q = idx * 4;
        float4 v;
        { const int qq = q + 0, d = qq / 3; v.x = row[d + (qq - 3 * d)]; }
        { const int qq = q + 1, d = qq / 3; v.y = row[d + (qq - 3 * d)]; }
        { const int qq = q + 2, d = qq / 3; v.z = row[d + (qq - 3 * d)]; }
        { const int qq = q + 3, d = qq / 3; v.w = row[d + (qq - 3 * d)]; }
        orow[idx] = v;
      }
    }
  }
}

extern "C" void kernel_launch(void* const* d_in, const int* in_sizes, int n_in,
                              void* d_out, int out_size, void* d_ws, size_t ws_size,
                              hipStream_t stream) {
  const float* x  = (const float*)d_in[0];  // [8,3,512,512] fp32
  float* out      = (float*)d_out;          // [8,3,1536,1536] fp32
  (void)in_sizes; (void)n_in; (void)out_size; (void)d_ws; (void)ws_size;

  dim3 grid(PAD_W, BC);   // 514 x 24 blocks, one per (padded-row, b*c)
  unfold_rows_kernel<<<grid, BDIM, 0, stream>>>(x, out);
}